// GCN_7687991459994
// MI455X (gfx1250) — compile-verified
//
#include <hip/hip_runtime.h>

typedef __attribute__((ext_vector_type(2))) float v2f;
typedef __attribute__((ext_vector_type(8))) float v8f;

#define BN_EPS 1e-5f

// ---------------------------------------------------------------- utilities
__global__ void zero_f32(float* __restrict__ p, long long n) {
  long long i = (long long)blockIdx.x * blockDim.x + threadIdx.x;
  if (i < n) p[i] = 0.0f;
}

__global__ void count_deg(const int* __restrict__ dst, float* __restrict__ deg, int e) {
  int i = blockIdx.x * blockDim.x + threadIdx.x;
  if (i < e) atomicAdd(&deg[dst[i]], 1.0f);
}

__global__ void make_dinv(float* __restrict__ deg, int n) {
  int i = blockIdx.x * blockDim.x + threadIdx.x;
  if (i < n) deg[i] = rsqrtf(deg[i] + 1.0f);   // includes self-loop (+1)
}

// ------------------------------------------------- fp32 WMMA GEMM  C = A*B
// A:[M,K] row-major, B:[K,NC] row-major, C:[M,NC]. One wave per 16x16 tile.
// Block covers MT m-tiles x (NC/16) n-tiles; grid.x * MT * 16 == M exactly.
// Two independent accumulator chains (even/odd K-steps) to double WMMA ILP.
template <int K, int NC, int MT>
__global__ void __launch_bounds__(MT*(NC/16)*32)
gemm_wmma_f32(const float* __restrict__ A, const float* __restrict__ B,
              float* __restrict__ C) {
  constexpr int KP  = K + 4;   // pad: A-frag reads stride 4 banks/row -> conflict-free
  constexpr int NCP = NC + 8;  // pad: B-frag lane-halves hit disjoint bank ranges
  constexpr int NW  = MT * (NC/16);
  __shared__ float sA[MT*16*KP];
  __shared__ float sB[K*NCP];

  const int tid = threadIdx.x;
  const int nthreads = NW * 32;
  const float* Ablk = A + (size_t)blockIdx.x * (MT*16) * K;

  for (int idx = tid; idx < MT*16*(K/4); idx += nthreads) {
    int r = idx / (K/4), c4 = (idx % (K/4)) * 4;
    *(float4*)&sA[r*KP + c4] = *(const float4*)&Ablk[(size_t)r*K + c4];
  }
  for (int idx = tid; idx < K*(NC/4); idx += nthreads) {
    int r = idx / (NC/4), c4 = (idx % (NC/4)) * 4;
    *(float4*)&sB[r*NCP + c4] = *(const float4*)&B[(size_t)r*NC + c4];
  }
  __syncthreads();

  const int wave  = tid >> 5, lane = tid & 31;
  const int mt    = wave / (NC/16), nt = wave % (NC/16);
  const int rrow  = lane & 15;      // A: tile row (M);  B: tile col (N)
  const int khalf = lane >> 4;      // 0 -> K{0,1}, 1 -> K{2,3}  (ISA 16x4/4x16 layout)

  const float* a_ptr = &sA[(mt*16 + rrow)*KP + khalf*2];
  const float* b_ptr = &sB[(khalf*2)*NCP + nt*16 + rrow];

  v8f c0 = {};
  v8f c1 = {};
  #pragma unroll 4
  for (int ks = 0; ks < K/8; ++ks) {           // two 16x16x4 steps per iter
    v2f a0; a0.x = a_ptr[0];     a0.y = a_ptr[1];
    v2f b0; b0.x = b_ptr[0];     b0.y = b_ptr[NCP];
    v2f a1; a1.x = a_ptr[4];     a1.y = a_ptr[5];
    v2f b1; b1.x = b_ptr[4*NCP]; b1.y = b_ptr[5*NCP];
    c0 = __builtin_amdgcn_wmma_f32_16x16x4_f32(false, a0, false, b0,
                                               (short)0, c0, false, false);
    c1 = __builtin_amdgcn_wmma_f32_16x16x4_f32(false, a1, false, b1,
                                               (short)0, c1, false, false);
    a_ptr += 8;
    b_ptr += 8*NCP;
  }
  v8f c = c0 + c1;

  // C/D layout: VGPR r -> M = r (lanes 0-15) / 8+r (lanes 16-31), N = lane%16
  const int gm  = blockIdx.x*(MT*16) + mt*16 + khalf*8;
  const int col = nt*16 + rrow;
  #pragma unroll
  for (int r = 0; r < 8; ++r)
    C[(size_t)(gm + r)*NC + col] = c[r];
}

// ------------------------------------------------- edge scatter (aggregate)
// F/2 consecutive threads per edge: broadcast src/dst/dinv loads, coalesced
// float2 gather of h[src], cacheline-grouped f32 atomics into agg[dst].
template <int F>
__global__ void scatter_edges(const int* __restrict__ src, const int* __restrict__ dst,
                              const float* __restrict__ dinv, const float* __restrict__ h,
                              float* __restrict__ agg, long long total) {
  long long t = (long long)blockIdx.x * blockDim.x + threadIdx.x;
  if (t >= total) return;
  constexpr int HALF = F / 2;
  int e = (int)(t / HALF);
  int f = (int)(t % HALF) * 2;
  int s = src[e], d = dst[e];
  float norm = dinv[s] * dinv[d];
  float2 hv = *(const float2*)&h[(size_t)s*F + f];
  atomicAdd(&agg[(size_t)d*F + f],     hv.x * norm);
  atomicAdd(&agg[(size_t)d*F + f + 1], hv.y * norm);
}

// --------------------------------- self-loop + bias + BatchNorm(eval) + ReLU
template <int F>
__global__ void finalize_bn(float* __restrict__ agg, const float* __restrict__ h,
                            const float* __restrict__ dinv, const float* __restrict__ bias,
                            const float* __restrict__ gamma, const float* __restrict__ beta,
                            const float* __restrict__ mean, const float* __restrict__ var,
                            long long total) {
  long long t = (long long)blockIdx.x * blockDim.x + threadIdx.x;
  if (t >= total) return;
  int node = (int)(t / F), f = (int)(t % F);
  float di = dinv[node];
  float v = agg[t] + h[t] * di * di + bias[f];
  v = (v - mean[f]) * rsqrtf(var[f] + BN_EPS) * gamma[f] + beta[f];
  agg[t] = fmaxf(v, 0.0f);   // in place: agg becomes the layer activation
}

// --------------------------------- layer-2 finalize fused with linear head
__global__ void final_head(const float* __restrict__ agg2, const float* __restrict__ h2,
                           const float* __restrict__ dinv, const float* __restrict__ b2,
                           const float* __restrict__ g2, const float* __restrict__ be2,
                           const float* __restrict__ m2, const float* __restrict__ v2,
                           const float* __restrict__ Wl, const float* __restrict__ bl,
                           float* __restrict__ out, int n) {
  int node = blockIdx.x * blockDim.x + threadIdx.x;
  if (node >= n) return;
  float di = dinv[node];
  float di2 = di * di;
  float acc0 = bl[0], acc1 = bl[1];
  const float* ar = agg2 + (size_t)node * 32;
  const float* hr = h2 + (size_t)node * 32;
  #pragma unroll
  for (int f = 0; f < 32; ++f) {
    float v = ar[f] + hr[f] * di2 + b2[f];
    v = (v - m2[f]) * rsqrtf(v2[f] + BN_EPS) * g2[f] + be2[f];
    v = fmaxf(v, 0.0f);
    acc0 += v * Wl[f*2 + 0];
    acc1 += v * Wl[f*2 + 1];
  }
  out[(size_t)node*2 + 0] = acc0;
  out[(size_t)node*2 + 1] = acc1;
}

// ---------------------------------------------------------------- launcher
extern "C" void kernel_launch(void* const* d_in, const int* in_sizes, int n_in,
                              void* d_out, int out_size, void* d_ws, size_t ws_size,
                              hipStream_t stream) {
  const float* x      = (const float*)d_in[0];
  const int*   ei     = (const int*)  d_in[1];
  const float* W1     = (const float*)d_in[2];
  const float* b1     = (const float*)d_in[3];
  const float* gamma1 = (const float*)d_in[4];
  const float* beta1  = (const float*)d_in[5];
  const float* mean1  = (const float*)d_in[6];
  const float* var1   = (const float*)d_in[7];
  const float* W2     = (const float*)d_in[8];
  const float* b2     = (const float*)d_in[9];
  const float* gamma2 = (const float*)d_in[10];
  const float* beta2  = (const float*)d_in[11];
  const float* mean2  = (const float*)d_in[12];
  const float* var2   = (const float*)d_in[13];
  const float* Wl     = (const float*)d_in[14];
  const float* bl     = (const float*)d_in[15];

  const int n = in_sizes[0] / 256;   // N nodes (100000)
  const int e = in_sizes[1] / 2;     // E edges (3200000)
  const int* src = ei;
  const int* dst = ei + e;

  float* ws   = (float*)d_ws;
  float* dinv = ws;                       // [n]
  float* h1   = dinv + (size_t)n;         // [n,64]
  float* agg1 = h1   + (size_t)n * 64;    // [n,64] -> becomes relu(bn1) act
  float* h2   = agg1 + (size_t)n * 64;    // [n,32]
  float* agg2 = h2   + (size_t)n * 32;    // [n,32]

  const int T = 256;
  auto blocks = [](long long total, int t) {
    return (unsigned int)((total + t - 1) / t);
  };

  // zero accumulators (re-done every call: deterministic)
  zero_f32<<<blocks(n, T), T, 0, stream>>>(dinv, n);
  zero_f32<<<blocks((long long)n * 64, T), T, 0, stream>>>(agg1, (long long)n * 64);
  zero_f32<<<blocks((long long)n * 32, T), T, 0, stream>>>(agg2, (long long)n * 32);

  // degree -> dinv = rsqrt(deg+1)
  count_deg<<<blocks(e, T), T, 0, stream>>>(dst, dinv, e);
  make_dinv<<<blocks(n, T), T, 0, stream>>>(dinv, n);

  // layer 1: h1 = x @ W1  (WMMA f32), N divisible by 16
  gemm_wmma_f32<256, 64, 1><<<n / 16, 128, 0, stream>>>(x, W1, h1);

  // edge aggregation 1
  scatter_edges<64><<<blocks((long long)e * 32, T), T, 0, stream>>>(
      src, dst, dinv, h1, agg1, (long long)e * 32);

  // self-loop + bias + BN + ReLU (agg1 -> activation, in place)
  finalize_bn<64><<<blocks((long long)n * 64, T), T, 0, stream>>>(
      agg1, h1, dinv, b1, gamma1, beta1, mean1, var1, (long long)n * 64);

  // layer 2: h2 = act1 @ W2 (WMMA f32), N divisible by 32
  gemm_wmma_f32<64, 32, 2><<<n / 32, 128, 0, stream>>>(agg1, W2, h2);

  // edge aggregation 2
  scatter_edges<32><<<blocks((long long)e * 16, T), T, 0, stream>>>(
      src, dst, dinv, h2, agg2, (long long)e * 16);

  // finalize layer 2 + linear head
  final_head<<<blocks(n, T), T, 0, stream>>>(
      agg2, h2, dinv, b2, gamma2, beta2, mean2, var2, Wl, bl,
      (float*)d_out, n);
}